// JointGraphEncoder_25993142075735
// MI455X (gfx1250) — compile-verified
//
#include <hip/hip_runtime.h>
#include <hip/hip_bf16.h>

// ---------------------------------------------------------------------------
// JointGraphEncoder for MI455X (gfx1250, wave32, WMMA).
//
// Problem sizes (from reference):
//   N_NODES=50000, N_EDGES=800000, NUM_GRAPHS=512, NODE_F=HID=64, EDGE_F=16,
//   OUT=128, 3 GINE layers. Output: [512,128] f32.
//
// Memory-bound analysis: ~1.5 GB of HBM traffic (edge gather/scatter dominates)
// vs ~8 GFLOP -> pure bandwidth problem at 23.3 TB/s. We therefore:
//   * fuse edge-linear + message + relu + segment-sum scatter (never write e)
//   * use V_WMMA_F32_16X16X4_F32 for all big matmuls (f32-exact, matrix pipe)
//   * use hardware global_atomic_add_f32 for the scatter adds
//
// Input flattening assumption (d_in order):
//   top-level dict insertion order: x, edge_attr, edge_index, batch, ge, params
//   params flattened as a JAX pytree (dict keys sorted alphabetically):
//     convs[i]:   bn.b, bn.g, bn.mean, bn.var, lin1.W, lin1.b, lin2.W, lin2.b
//     edge_lins[i]: W, b
//     enc:        [0].W, [0].b, [1].W, [1].b
//     ge:         bn.b, bn.g, bn.mean, bn.var, lin1.W, lin1.b, lin2.W, lin2.b
//     out_lin:    [0].W, [0].b, [1].W, [1].b
// ---------------------------------------------------------------------------

#define N_NODES   50000
#define N_EDGES   800000
#define NUM_G     512
#define NODE_F    64
#define EDGE_F    16
#define HID       64
#define OUTF      128
#define BN_EPS    1e-5f
#define LEAKY     0.01f

typedef float v2f __attribute__((ext_vector_type(2)));
typedef float v8f __attribute__((ext_vector_type(8)));

// D(16x16,f32) = A(16x4,f32) * B(4x16,f32) + C
__device__ __forceinline__ v8f wmma4(v2f a, v2f b, v8f c) {
  return __builtin_amdgcn_wmma_f32_16x16x4_f32(false, a, false, b, (short)0, c,
                                               false, false);
}

// ---------------------------------------------------------------------------
// zero fill
// ---------------------------------------------------------------------------
__global__ void zero_kernel(float* __restrict__ p, long long n) {
  long long i = (long long)blockIdx.x * blockDim.x + threadIdx.x;
  long long stride = (long long)gridDim.x * blockDim.x;
  for (; i < n; i += stride) p[i] = 0.0f;
}

// ---------------------------------------------------------------------------
// Fused GINE edge stage, one wave (32 lanes) per 16-edge tile:
//   e   = edge_attr[16x16] @ We[16x64] + be           (16 x WMMA f32 16x16x4)
//   m   = relu(x[src] + e)
//   agg[dst] += m                                      (global_atomic_add_f32)
// A layout: lane L holds row M=L%16; VGPR0/1 hold K = koff, koff+1 with
//           koff = 2*(L/16)  (per ISA 7.12.2 "32-bit A-Matrix 16x4").
// B layout: lane L holds col N=L%16; VGPR0/1 hold same K rows.
// C/D layout: VGPR r, lane L -> element (M = r + 8*(L/16), N = L%16).
// ---------------------------------------------------------------------------
__global__ __launch_bounds__(32)
void edge_msg_kernel(const float* __restrict__ edge_attr,  // [E,16]
                     const int*   __restrict__ src,        // [E]
                     const int*   __restrict__ dst,        // [E]
                     const float* __restrict__ x,          // [N,64]
                     const float* __restrict__ We,         // [16,64]
                     const float* __restrict__ be,         // [64]
                     float*       __restrict__ agg,        // [N,64]
                     int numTiles) {
  const int tile = blockIdx.x;
  if (tile >= numTiles) return;
  const int lane = threadIdx.x;
  const int half = lane >> 4;
  const int l16  = lane & 15;
  const int koff = half * 2;

  __shared__ int sh_src[16];
  __shared__ int sh_dst[16];
  if (lane < 16) {
    sh_src[lane] = src[tile * 16 + lane];
    sh_dst[lane] = dst[tile * 16 + lane];
  }
  __syncthreads();

  // speculative prefetch of next tile's edge features (global_prefetch_b8)
  __builtin_prefetch(edge_attr + (size_t)(tile + 1) * 16 * EDGE_F, 0, 0);

  const float* arow = edge_attr + (size_t)(tile * 16 + l16) * EDGE_F;

  v8f acc[4] = {};
#pragma unroll
  for (int s = 0; s < 4; ++s) {
    const int k = 4 * s + koff;
    v2f a;
    a.x = arow[k];
    a.y = arow[k + 1];
    const float* w0 = We + (size_t)k * 64;
    const float* w1 = We + (size_t)(k + 1) * 64;
#pragma unroll
    for (int t = 0; t < 4; ++t) {
      v2f b;
      b.x = w0[t * 16 + l16];
      b.y = w1[t * 16 + l16];
      acc[t] = wmma4(a, b, acc[t]);
    }
  }

  float bias[4];
#pragma unroll
  for (int t = 0; t < 4; ++t) bias[t] = be[t * 16 + l16];

#pragma unroll
  for (int r = 0; r < 8; ++r) {
    const int m    = r + 8 * half;
    const int srow = sh_src[m];
    const int drow = sh_dst[m];
    const float* xr = x   + (size_t)srow * 64;
    float*       ar = agg + (size_t)drow * 64;
#pragma unroll
    for (int t = 0; t < 4; ++t) {
      const int n = t * 16 + l16;
      float v = acc[t][r] + bias[t] + xr[n];
      v = fmaxf(v, 0.0f);
      unsafeAtomicAdd(ar + n, v);  // global_atomic_add_f32
    }
  }
}

// ---------------------------------------------------------------------------
// Fused GINE node stage, one wave per 16-node tile (rows contiguous):
//   h = x + agg ; h = relu(bn(h@W1 + b1)) ; x' = relu(h@W2 + b2)
// Both 64x64 GEMMs via 64 WMMA f32 16x16x4 each; LDS used to re-layout the
// intermediate from C/D layout back to A layout.
// ---------------------------------------------------------------------------
__global__ __launch_bounds__(32)
void node_update_kernel(const float* __restrict__ x,    // [N,64]
                        const float* __restrict__ agg,  // [N,64]
                        const float* __restrict__ W1,   // [64,64]
                        const float* __restrict__ b1,   // [64]
                        const float* __restrict__ bng,
                        const float* __restrict__ bnb,
                        const float* __restrict__ bnm,
                        const float* __restrict__ bnv,
                        const float* __restrict__ W2,   // [64,64]
                        const float* __restrict__ b2,   // [64]
                        float*       __restrict__ xout, // [N,64]
                        int numTiles) {
  __shared__ float sh_h[16 * 64];
  __shared__ float sh_t[16 * 64];

  const int tile = blockIdx.x;
  if (tile >= numTiles) return;
  const int lane = threadIdx.x;
  const int half = lane >> 4;
  const int l16  = lane & 15;
  const int koff = half * 2;
  const size_t base = (size_t)tile * 16 * 64;

  // h = x + agg : tile is 1024 contiguous floats, each lane moves 8x float4
#pragma unroll
  for (int j = 0; j < 8; ++j) {
    const int idx = j * 128 + lane * 4;
    const float4 xv = *(const float4*)(x + base + idx);
    const float4 av = *(const float4*)(agg + base + idx);
    float4 h;
    h.x = xv.x + av.x; h.y = xv.y + av.y;
    h.z = xv.z + av.z; h.w = xv.w + av.w;
    *(float4*)(sh_h + idx) = h;
  }
  __syncthreads();

  // ---- lin1 ----
  v8f acc[4] = {};
#pragma unroll
  for (int s = 0; s < 16; ++s) {
    const int k = 4 * s + koff;
    v2f a;
    a.x = sh_h[l16 * 64 + k];
    a.y = sh_h[l16 * 64 + k + 1];
    const float* w0 = W1 + (size_t)k * 64;
    const float* w1 = W1 + (size_t)(k + 1) * 64;
#pragma unroll
    for (int t = 0; t < 4; ++t) {
      v2f b;
      b.x = w0[t * 16 + l16];
      b.y = w1[t * 16 + l16];
      acc[t] = wmma4(a, b, acc[t]);
    }
  }

  // bias + batchnorm + relu, re-stage in row-major LDS tile
#pragma unroll
  for (int t = 0; t < 4; ++t) {
    const int n = t * 16 + l16;
    const float scale = rsqrtf(bnv[n] + BN_EPS) * bng[n];
    const float mean  = bnm[n];
    const float beta  = bnb[n];
    const float bias  = b1[n];
#pragma unroll
    for (int r = 0; r < 8; ++r) {
      const int m = r + 8 * half;
      float v = acc[t][r] + bias;
      v = (v - mean) * scale + beta;
      v = fmaxf(v, 0.0f);
      sh_t[m * 64 + n] = v;
    }
  }
  __syncthreads();

  // ---- lin2 ----
  v8f acc2[4] = {};
#pragma unroll
  for (int s = 0; s < 16; ++s) {
    const int k = 4 * s + koff;
    v2f a;
    a.x = sh_t[l16 * 64 + k];
    a.y = sh_t[l16 * 64 + k + 1];
    const float* w0 = W2 + (size_t)k * 64;
    const float* w1 = W2 + (size_t)(k + 1) * 64;
#pragma unroll
    for (int t = 0; t < 4; ++t) {
      v2f b;
      b.x = w0[t * 16 + l16];
      b.y = w1[t * 16 + l16];
      acc2[t] = wmma4(a, b, acc2[t]);
    }
  }

#pragma unroll
  for (int t = 0; t < 4; ++t) {
    const int n = t * 16 + l16;
    const float bias = b2[n];
#pragma unroll
    for (int r = 0; r < 8; ++r) {
      const int m = r + 8 * half;
      xout[base + m * 64 + n] = fmaxf(acc2[t][r] + bias, 0.0f);
    }
  }
}

// ---------------------------------------------------------------------------
// scatter-mean pooling over graphs
// ---------------------------------------------------------------------------
__global__ void pool_scatter_kernel(const float* __restrict__ x,
                                    const int* __restrict__ batch,
                                    float* __restrict__ sums,
                                    float* __restrict__ cnt) {
  const long long tid = (long long)blockIdx.x * blockDim.x + threadIdx.x;
  if (tid >= (long long)N_NODES * 64) return;
  const int node = (int)(tid >> 6);
  const int f    = (int)(tid & 63);
  const int g    = batch[node];
  unsafeAtomicAdd(&sums[g * 64 + f], x[tid]);
  if (f == 0) unsafeAtomicAdd(&cnt[g], 1.0f);
}

__global__ void pool_mean_kernel(const float* __restrict__ sums,
                                 const float* __restrict__ cnt,
                                 float* __restrict__ g) {
  const int tid = blockIdx.x * blockDim.x + threadIdx.x;
  if (tid >= NUM_G * 64) return;
  g[tid] = sums[tid] / fmaxf(cnt[tid >> 6], 1.0f);
}

// ---------------------------------------------------------------------------
// generic small dense layer: out = act(bn(A@W + b)); bn optional.
// act: 0=none, 1=relu, 2=leaky(0.01). Only used for the 512-row heads.
// ---------------------------------------------------------------------------
__global__ void dense_kernel(const float* __restrict__ A,
                             const float* __restrict__ W,
                             const float* __restrict__ bias,
                             const float* __restrict__ bng,
                             const float* __restrict__ bnb,
                             const float* __restrict__ bnm,
                             const float* __restrict__ bnv,
                             float* __restrict__ out,
                             int M, int K, int N, int act) {
  const int tid = blockIdx.x * blockDim.x + threadIdx.x;
  if (tid >= M * N) return;
  const int m = tid / N;
  const int n = tid - m * N;
  const float* a = A + (size_t)m * K;
  float acc = bias[n];
  for (int k = 0; k < K; ++k) acc += a[k] * W[(size_t)k * N + n];
  if (bng) acc = (acc - bnm[n]) * rsqrtf(bnv[n] + BN_EPS) * bng[n] + bnb[n];
  if (act == 1) acc = fmaxf(acc, 0.0f);
  else if (act == 2) acc = (acc > 0.0f) ? acc : LEAKY * acc;
  out[tid] = acc;
}

__global__ void concat_kernel(const float* __restrict__ a,
                              const float* __restrict__ b,
                              float* __restrict__ z, int M, int Na, int Nb) {
  const int tid = blockIdx.x * blockDim.x + threadIdx.x;
  const int Nt = Na + Nb;
  if (tid >= M * Nt) return;
  const int m = tid / Nt;
  const int c = tid - m * Nt;
  z[tid] = (c < Na) ? a[m * Na + c] : b[m * Nb + (c - Na)];
}

// ---------------------------------------------------------------------------
extern "C" void kernel_launch(void* const* d_in, const int* in_sizes, int n_in,
                              void* d_out, int out_size, void* d_ws,
                              size_t ws_size, hipStream_t stream) {
  (void)in_sizes; (void)n_in; (void)out_size; (void)ws_size;

  const float* x         = (const float*)d_in[0];
  const float* edge_attr = (const float*)d_in[1];
  const int*   eidx      = (const int*)d_in[2];
  const int*   batch     = (const int*)d_in[3];
  const float* ge        = (const float*)d_in[4];
  const int*   src       = eidx;
  const int*   dst       = eidx + N_EDGES;

  auto P = [&](int i) { return (const float*)d_in[i]; };
  // params leaf indices (see flattening assumption at top of file)
  // convs[i]: base 5+8i -> bn.b, bn.g, bn.mean, bn.var, l1.W, l1.b, l2.W, l2.b
  // edge_lins[i]: 29+2i = W, 30+2i = b
  // enc: 35..38 ; ge: 39..46 ; out_lin: 47..50

  float* ws    = (float*)d_ws;
  float* agg   = ws; ws += (size_t)N_NODES * 64;
  float* xA    = ws; ws += (size_t)N_NODES * 64;
  float* xB    = ws; ws += (size_t)N_NODES * 64;
  float* sums  = ws; ws += NUM_G * 64;
  float* cnt   = ws; ws += NUM_G;          // contiguous after sums (zeroed together)
  float* gmean = ws; ws += NUM_G * 64;
  float* gh1   = ws; ws += NUM_G * OUTF;
  float* gout  = ws; ws += NUM_G * OUTF;
  float* geh   = ws; ws += NUM_G * 64;
  float* geout = ws; ws += NUM_G * OUTF;
  float* zcat  = ws; ws += NUM_G * 2 * OUTF;
  float* ench  = ws; ws += NUM_G * 2 * OUTF;

  const int edgeTiles = N_EDGES / 16;   // 50000
  const int nodeTiles = N_NODES / 16;   // 3125

  for (int i = 0; i < 3; ++i) {
    const float* xin  = (i == 0) ? x : ((i == 1) ? xA : xB);
    float*       xout = (i == 0) ? xA : ((i == 1) ? xB : xA);
    zero_kernel<<<2048, 256, 0, stream>>>(agg, (long long)N_NODES * 64);
    edge_msg_kernel<<<edgeTiles, 32, 0, stream>>>(
        edge_attr, src, dst, xin, P(29 + 2 * i), P(30 + 2 * i), agg, edgeTiles);
    const int cb = 5 + 8 * i;
    node_update_kernel<<<nodeTiles, 32, 0, stream>>>(
        xin, agg, P(cb + 4), P(cb + 5), P(cb + 1), P(cb + 0), P(cb + 2),
        P(cb + 3), P(cb + 6), P(cb + 7), xout, nodeTiles);
  }
  const float* xfinal = xA;  // after layer 2

  zero_kernel<<<64, 256, 0, stream>>>(sums, NUM_G * 64 + NUM_G);
  pool_scatter_kernel<<<((long long)N_NODES * 64 + 255) / 256, 256, 0, stream>>>(
      xfinal, batch, sums, cnt);
  pool_mean_kernel<<<(NUM_G * 64 + 255) / 256, 256, 0, stream>>>(sums, cnt, gmean);

  auto dense = [&](const float* A, const float* W, const float* b,
                   const float* g_, const float* bb, const float* mm,
                   const float* vv, float* out, int M, int K, int N, int act) {
    dense_kernel<<<(M * N + 255) / 256, 256, 0, stream>>>(A, W, b, g_, bb, mm,
                                                          vv, out, M, K, N, act);
  };

  // graph head: relu(g@W0+b0) @ W1 + b1
  dense(gmean, P(47), P(48), nullptr, nullptr, nullptr, nullptr, gh1, NUM_G, 64, OUTF, 1);
  dense(gh1,   P(49), P(50), nullptr, nullptr, nullptr, nullptr, gout, NUM_G, OUTF, OUTF, 0);
  // ge encoder: leaky(bn(ge@W+b)) @ W2 + b2
  dense(ge,    P(43), P(44), P(40), P(39), P(41), P(42), geh, NUM_G, 64, 64, 2);
  dense(geh,   P(45), P(46), nullptr, nullptr, nullptr, nullptr, geout, NUM_G, 64, OUTF, 0);
  // concat + final encoder
  concat_kernel<<<(NUM_G * 2 * OUTF + 255) / 256, 256, 0, stream>>>(
      gout, geout, zcat, NUM_G, OUTF, OUTF);
  dense(zcat, P(35), P(36), nullptr, nullptr, nullptr, nullptr, ench, NUM_G, 2 * OUTF, 2 * OUTF, 2);
  dense(ench, P(37), P(38), nullptr, nullptr, nullptr, nullptr, (float*)d_out,
        NUM_G, 2 * OUTF, OUTF, 0);
}